// HarmonicOscillatorBank_12773232738628
// MI455X (gfx1250) — compile-verified
//
#include <hip/hip_runtime.h>
#include <math.h>

// Problem constants (match reference)
#define SRATE    16000
#define NOSC     101
#define T_LEN    64000
#define B_SZ     16
#define CHUNK    512                    // time samples per block
#define NCHUNK   (T_LEN / CHUNK)        // 125 chunks per batch
#define THREADS  128                    // 4 consecutive t per thread

#define TWO_PI_F   6.28318530717958647692f
#define INV_2PI_F  0.15915494309189533577f
#define W_SCALE    (TWO_PI_F / (float)SRATE)

// ---------- CDNA5 async global->LDS copy (ASYNCcnt path) ----------
__device__ __forceinline__ void async_copy_b128(unsigned lds_off, const float* gaddr) {
    // GV mode: per-lane 64-bit global address, per-lane 32-bit LDS byte address.
    asm volatile("global_load_async_to_lds_b128 %0, %1, off"
                 :: "v"(lds_off), "v"(gaddr)
                 : "memory");
}
__device__ __forceinline__ void wait_async_le3() { asm volatile("s_wait_asynccnt 0x3" ::: "memory"); }
__device__ __forceinline__ void wait_async_le2() { asm volatile("s_wait_asynccnt 0x2" ::: "memory"); }
__device__ __forceinline__ void wait_async_le1() { asm volatile("s_wait_asynccnt 0x1" ::: "memory"); }
__device__ __forceinline__ void wait_async_le0() { asm volatile("s_wait_asynccnt 0x0" ::: "memory"); }

// ---------- K1: per-chunk sum of omega ----------
__global__ __launch_bounds__(THREADS)
void k_chunksum(const float* __restrict__ f0, float* __restrict__ chunk_sum) {
    __shared__ float red[THREADS];
    const int b   = blockIdx.x / NCHUNK;
    const int c   = blockIdx.x % NCHUNK;
    const int tid = threadIdx.x;
    const int t0  = c * CHUNK + tid * 4;

    float4 uu = *(const float4*)(f0 + (size_t)b * T_LEN + t0);
    float u[4] = {uu.x, uu.y, uu.z, uu.w};
    float s = 0.f;
#pragma unroll
    for (int i = 0; i < 4; ++i)
        s += W_SCALE * fmaf(u[i], 450.f, 50.f);

    red[tid] = s;
    __syncthreads();
#pragma unroll
    for (int d = THREADS / 2; d > 0; d >>= 1) {
        if (tid < d) red[tid] += red[tid + d];
        __syncthreads();
    }
    if (tid == 0) chunk_sum[blockIdx.x] = red[0];
}

// ---------- K2: exclusive scan of chunk sums (f64), reduced mod 2*pi ----------
__global__ void k_scan(const float* __restrict__ chunk_sum, float* __restrict__ chunk_base) {
    const int b = threadIdx.x;
    if (b < B_SZ) {
        double run = 0.0;
        for (int c = 0; c < NCHUNK; ++c) {
            chunk_base[b * NCHUNK + c] = (float)fmod(run, 6.283185307179586476);
            run += (double)chunk_sum[b * NCHUNK + c];
        }
    }
}

// One harmonic step: masked accumulate + Chebyshev advance (4 samples / thread)
#define CONSUME(a4)                                                  \
    do {                                                             \
        float av_[4] = {(a4).x, (a4).y, (a4).z, (a4).w};             \
        _Pragma("unroll")                                            \
        for (int i_ = 0; i_ < 4; ++i_) {                             \
            float am_ = (fh <= fnmax[i_]) ? av_[i_] : 0.f;           \
            acc[i_] = fmaf(am_, cc[i_], acc[i_]);                    \
            float nx_ = fmaf(x2[i_], cc[i_], -cp[i_]);               \
            cp[i_] = cc[i_];                                         \
            cc[i_] = nx_;                                            \
        }                                                            \
    } while (0)

// ---------- K3: phase recovery + 101-harmonic synthesis ----------
__global__ __launch_bounds__(THREADS)
void k_synth(const float* __restrict__ f0, const float* __restrict__ amp,
             const float* __restrict__ chunk_base, float* __restrict__ out) {
    __shared__ float4 lbuf[4][THREADS];   // 4-deep async pipeline: 16B per lane per buffer
    __shared__ float  scan[THREADS];

    const int blk = blockIdx.x;
    const int b   = blk / NCHUNK;
    const int c   = blk % NCHUNK;
    const int tid = threadIdx.x;
    const int t0  = c * CHUNK + tid * 4;

    // --- fundamental omega for my 4 samples ---
    float4 uu = *(const float4*)(f0 + (size_t)b * T_LEN + t0);
    float u[4] = {uu.x, uu.y, uu.z, uu.w};
    float fs[4], w[4];
#pragma unroll
    for (int i = 0; i < 4; ++i) {
        fs[i] = fmaf(u[i], 450.f, 50.f);     // Hz, in [50,500]
        w[i]  = W_SCALE * fs[i];             // rad/sample
    }
    // in-thread inclusive prefix
    float s[4];
    s[0] = w[0]; s[1] = s[0] + w[1]; s[2] = s[1] + w[2]; s[3] = s[2] + w[3];

    // --- cross-thread inclusive scan of per-thread sums (Hillis-Steele, once per block) ---
    scan[tid] = s[3];
    __syncthreads();
#pragma unroll
    for (int d = 1; d < THREADS; d <<= 1) {
        float v   = scan[tid];
        float add = (tid >= d) ? scan[tid - d] : 0.f;
        __syncthreads();
        scan[tid] = v + add;
        __syncthreads();
    }
    const float base = chunk_base[blk] + (scan[tid] - s[3]);  // exclusive prefix + chunk base (mod 2pi)

    // --- fundamental phase, reduced mod 2*pi (exact for integer harmonics) ---
    float x2[4], cc[4], cp[4], acc[4], fnmax[4];
#pragma unroll
    for (int i = 0; i < 4; ++i) {
        float ph = base + s[i];
        ph -= TWO_PI_F * truncf(ph * INV_2PI_F);   // ph >= 0, so trunc == floor
        float x = __cosf(ph);                      // single v_cos_f32 per sample
        x2[i] = x + x;
        cc[i] = x;                                 // cos(1*phi)
        cp[i] = 1.f;                               // cos(0*phi)
        acc[i] = 0.f;
        fnmax[i] = 8000.f / fs[i];                 // anti-alias cutoff: h <= 8000/f
    }

    // --- 101-harmonic amp stream: 4-deep async-LDS pipeline, fully static indices ---
    const unsigned off0 = (unsigned)(size_t)(&lbuf[0][tid]);  // low 32 bits of flat addr = LDS offset
    const unsigned BUFB = (unsigned)(THREADS * sizeof(float4)); // 2048B between buffers
    const float* grow = amp + ((size_t)b * NOSC) * T_LEN + t0;  // row of harmonic h=1

    // prologue: issue h = 1..4 into buffers 0..3
    async_copy_b128(off0 + 0 * BUFB, grow + 0 * (size_t)T_LEN);
    async_copy_b128(off0 + 1 * BUFB, grow + 1 * (size_t)T_LEN);
    async_copy_b128(off0 + 2 * BUFB, grow + 2 * (size_t)T_LEN);
    async_copy_b128(off0 + 3 * BUFB, grow + 3 * (size_t)T_LEN);

    float fh = 1.f;
    // main loop: h = 1..96 in groups of 4; each step consumes h then refills its buffer with h+4
    for (int g = 0; g < 24; ++g) {
#pragma unroll
        for (int j = 0; j < 4; ++j) {
            wait_async_le3();                          // oldest (h) complete
            float4 a4 = lbuf[j][tid];                  // lane-private 16B: no barrier needed
            CONSUME(a4);                               // data consumed before refill is issued
            async_copy_b128(off0 + (unsigned)j * BUFB, grow + (size_t)(4 + j) * T_LEN);
            fh += 1.f;
        }
        grow += 4 * (size_t)T_LEN;
    }
    // tail: h = 97..101 (grow now points at row 97)
    {
        wait_async_le3();                              // h=97 done (outstanding 97..100)
        float4 a4 = lbuf[0][tid]; CONSUME(a4);
        async_copy_b128(off0, grow + 4 * (size_t)T_LEN);  // issue h=101 -> buf0
        fh += 1.f;
        wait_async_le3();                              // h=98 done (outstanding 98..101)
        a4 = lbuf[1][tid]; CONSUME(a4); fh += 1.f;
        wait_async_le2();                              // h=99 done
        a4 = lbuf[2][tid]; CONSUME(a4); fh += 1.f;
        wait_async_le1();                              // h=100 done
        a4 = lbuf[3][tid]; CONSUME(a4); fh += 1.f;
        wait_async_le0();                              // h=101 done
        a4 = lbuf[0][tid]; CONSUME(a4);
    }

    *(float4*)(out + (size_t)b * T_LEN + t0) = make_float4(acc[0], acc[1], acc[2], acc[3]);
}

extern "C" void kernel_launch(void* const* d_in, const int* in_sizes, int n_in,
                              void* d_out, int out_size, void* d_ws, size_t ws_size,
                              hipStream_t stream) {
    (void)in_sizes; (void)n_in; (void)out_size; (void)ws_size;
    const float* f0  = (const float*)d_in[0];   // [B,1,T]
    const float* amp = (const float*)d_in[1];   // [B,NOSC,T]
    float* out = (float*)d_out;                 // [B,T]

    float* chunk_sum  = (float*)d_ws;                         // [B*NCHUNK]
    float* chunk_base = chunk_sum + (size_t)B_SZ * NCHUNK;    // [B*NCHUNK]

    k_chunksum<<<B_SZ * NCHUNK, THREADS, 0, stream>>>(f0, chunk_sum);
    k_scan    <<<1, 32, 0, stream>>>(chunk_sum, chunk_base);
    k_synth   <<<B_SZ * NCHUNK, THREADS, 0, stream>>>(f0, amp, chunk_base, out);
}